// DELLoss_13005160973096
// MI455X (gfx1250) — compile-verified
//
#include <hip/hip_runtime.h>

// DEL loss: per-class segment stats over z[N,64] (bandwidth-bound, ~528MB @ 23.3TB/s),
// then 128x128x64 f32 Gram via V_WMMA_F32_16X16X4_F32 for pairwise center distances.

typedef __attribute__((ext_vector_type(2))) float v2f;
typedef __attribute__((ext_vector_type(8))) float v8f;

#define KCLS 128
#define DIM  64
// ws float layout: [0,128) counts | [128, 128+8192) sums | [8320, 8448) sumsq
#define WS_COUNTS 0
#define WS_SUMS   KCLS
#define WS_SUMSQ  (KCLS + KCLS * DIM)
#define WS_FLOATS (KCLS + KCLS * DIM + KCLS)

__global__ void del_zero_ws(float* __restrict__ ws, int n) {
    int i = blockIdx.x * blockDim.x + threadIdx.x;
    if (i < n) ws[i] = 0.0f;
}

// Streaming pass: LDS-resident per-class accumulators, one global flush per block.
__global__ void del_accum(const float* __restrict__ z,
                          const long long* __restrict__ labels,
                          float* __restrict__ ws,
                          int nrows, int rowsPerBlock) {
    __shared__ float s_sums[KCLS * DIM];   // 32 KB
    __shared__ float s_counts[KCLS];
    __shared__ float s_sumsq[KCLS];

    const int tid = threadIdx.x;
    for (int i = tid; i < KCLS * DIM; i += blockDim.x) s_sums[i] = 0.0f;
    if (tid < KCLS) { s_counts[tid] = 0.0f; s_sumsq[tid] = 0.0f; }
    __syncthreads();

    const int sg = tid >> 4;          // 16-lane subgroup id (0..15), one row each
    const int ln = tid & 15;          // lane within subgroup: owns dims ln*4..ln*4+3
    const int wl = tid & 31;          // lane within wave32

    long long rowStart = (long long)blockIdx.x * (long long)rowsPerBlock;
    long long rowEnd   = rowStart + rowsPerBlock;
    if (rowEnd > nrows) rowEnd = nrows;

    for (long long base = rowStart; base < rowEnd; base += 16) {
        long long r = base + sg;
        bool active = (r < rowEnd);

        int lbl = -1;
        if (active && ln == 0) {
            long long lb = labels[r];
            lbl = (lb >= 0 && lb < KCLS) ? (int)lb : -1;
        }
        // broadcast label from lane 0 of each 16-lane half of the wave
        lbl = __shfl(lbl, wl & 16, 32);

        if (active && lbl >= 0) {
            const float4* zp = (const float4*)(z + r * (long long)DIM);
            float4 v = zp[ln];                       // wave reads 512 contiguous bytes
            float* row = &s_sums[lbl * DIM + ln * 4];
            atomicAdd(&row[0], v.x);
            atomicAdd(&row[1], v.y);
            atomicAdd(&row[2], v.z);
            atomicAdd(&row[3], v.w);
            float sq = v.x * v.x + v.y * v.y + v.z * v.z + v.w * v.w;
            sq += __shfl_xor(sq, 8, 32);             // reduce within 16-lane subgroup
            sq += __shfl_xor(sq, 4, 32);
            sq += __shfl_xor(sq, 2, 32);
            sq += __shfl_xor(sq, 1, 32);
            if (ln == 0) {
                atomicAdd(&s_sumsq[lbl], sq);
                atomicAdd(&s_counts[lbl], 1.0f);
            }
        }
    }
    __syncthreads();

    for (int i = tid; i < KCLS; i += blockDim.x) {
        atomicAdd(&ws[WS_COUNTS + i], s_counts[i]);
        atomicAdd(&ws[WS_SUMSQ + i], s_sumsq[i]);
    }
    for (int i = tid; i < KCLS * DIM; i += blockDim.x) {
        atomicAdd(&ws[WS_SUMS + i], s_sums[i]);
    }
}

// Single-block finalize: centers, variances, then C*C^T with V_WMMA_F32_16X16X4_F32.
__global__ void del_finalize(const float* __restrict__ ws, float* __restrict__ out) {
    __shared__ float cen[KCLS * DIM];   // 32 KB
    __shared__ float sqn[KCLS];
    __shared__ int   validf[KCLS];
    __shared__ float s_scal[4];         // 0: sum_var(has2), 1: n_has2, 2: dist_sum, 3: n_pairs
    __shared__ int   s_nvalid;

    const int tid = threadIdx.x;
    if (tid == 0) {
        s_scal[0] = s_scal[1] = s_scal[2] = s_scal[3] = 0.0f;
        s_nvalid = 0;
    }
    __syncthreads();

    if (tid < KCLS) {
        float n    = ws[WS_COUNTS + tid];
        float safe = fmaxf(n, 1.0f);
        float sn   = 0.0f;
        for (int d = 0; d < DIM; ++d) {
            float c = ws[WS_SUMS + tid * DIM + d] / safe;
            cen[tid * DIM + d] = c;
            sn += c * c;
        }
        sqn[tid]    = sn;
        int valid   = (n > 0.0f);
        validf[tid] = valid;
        if (valid) atomicAdd(&s_nvalid, 1);
        if (n > 1.0f) {
            float var = ws[WS_SUMSQ + tid] / safe - sn;
            atomicAdd(&s_scal[0], var);
            atomicAdd(&s_scal[1], 1.0f);
        }
    }
    __syncthreads();

    // --- Gram = cen * cen^T via f32 WMMA; 8 waves x 8 tiles = 64 tiles of 16x16.
    // Uniform control flow -> EXEC all-1s for every WMMA issue.
    const int wave = tid >> 5;
    const int lane = tid & 31;
    const int m    = lane & 15;            // A: row-in-tile; B/D: column-in-tile (N)
    const int koff = (lane >> 4) << 1;     // K sub-offset: lanes 0-15 -> {0,1}, 16-31 -> {2,3}
    const int mbase = (lane >> 4) << 3;    // D-matrix: VGPR v -> M = v + mbase

    float dist_sum = 0.0f;
    float pair_cnt = 0.0f;

    for (int t = 0; t < 8; ++t) {
        int tile = wave * 8 + t;
        int ti = tile >> 3;
        int tj = tile & 7;
        const float* arow = &cen[(ti * 16 + m) * DIM];
        const float* brow = &cen[(tj * 16 + m) * DIM];

        v8f acc = {};
        for (int kb = 0; kb < DIM; kb += 4) {
            v2f a, b;
            a.x = arow[kb + koff];
            a.y = arow[kb + koff + 1];
            b.x = brow[kb + koff];
            b.y = brow[kb + koff + 1];
            acc = __builtin_amdgcn_wmma_f32_16x16x4_f32(
                /*neg_a=*/false, a, /*neg_b=*/false, b,
                /*c_mod=*/(short)0, acc, /*reuse_a=*/false, /*reuse_b=*/false);
        }

        // acc[v] = G[ti*16 + mbase + v][tj*16 + m]; keep i<j, both valid.
        for (int v = 0; v < 8; ++v) {
            int i = ti * 16 + mbase + v;
            int j = tj * 16 + m;
            if (i < j && validf[i] && validf[j]) {
                float d2 = sqn[i] + sqn[j] - 2.0f * acc[v];
                dist_sum += sqrtf(fmaxf(d2, 0.0f));
                pair_cnt += 1.0f;
            }
        }
    }
    atomicAdd(&s_scal[2], dist_sum);
    atomicAdd(&s_scal[3], pair_cnt);
    __syncthreads();

    if (tid == 0) {
        float intra = (s_scal[1] > 0.0f) ? (s_scal[0] / fmaxf(s_scal[1], 1.0f)) : 0.0f;
        float inter = (s_scal[3] > 0.0f) ? (-s_scal[2] / fmaxf(s_scal[3], 1.0f)) : 0.0f;
        float loss  = intra + 0.5f * inter;
        out[0] = (s_nvalid >= 2) ? loss : 0.0f;
    }
}

extern "C" void kernel_launch(void* const* d_in, const int* in_sizes, int n_in,
                              void* d_out, int out_size, void* d_ws, size_t ws_size,
                              hipStream_t stream) {
    const float*     z      = (const float*)d_in[0];
    const long long* labels = (const long long*)d_in[1];   // reference dtype int64
    float* out = (float*)d_out;
    float* ws  = (float*)d_ws;

    const int N = in_sizes[1];   // label count == row count

    del_zero_ws<<<(WS_FLOATS + 255) / 256, 256, 0, stream>>>(ws, WS_FLOATS);

    const int nblocks = 1024;    // ~2K rows (~0.5 MB stream) per block
    const int rowsPerBlock = (N + nblocks - 1) / nblocks;
    del_accum<<<nblocks, 256, 0, stream>>>(z, labels, ws, N, rowsPerBlock);

    del_finalize<<<1, 256, 0, stream>>>(ws, out);
}